// AttnPool1D_48034914239255
// MI455X (gfx1250) — compile-verified
//
#include <hip/hip_runtime.h>
#include <cstdint>

typedef __attribute__((ext_vector_type(2))) float v2f;
typedef __attribute__((ext_vector_type(8))) float v8f;

#define B_SZ 32
#define T_SZ 4096
#define D_SZ 1024
#define CH 16                          // chunks per batch row -> 512 blocks
#define ROWS_PER_CHUNK (T_SZ / CH)     // 256
#define WAVES 8
#define THREADS 256
#define DSLICE (D_SZ / WAVES)          // 128 d-columns owned per wave
#define CTILES (DSLICE / 16)           // 8 WMMA column tiles per wave
#define ROWSTRIDE 1032                 // LDS row stride (floats); kills bank conflicts
#define XBUF_FLOATS (16 * ROWSTRIDE)   // one 16-row tile buffer
#define PART_STRIDE 1032               // per-(b,chunk) partial: 1024 acc + m + s + pad

__global__ __launch_bounds__(THREADS)
void attnpool_partial(const float* __restrict__ x,
                      const unsigned char* __restrict__ mask,
                      const float* __restrict__ query,
                      float* __restrict__ ws)
{
    const int b     = blockIdx.y;
    const int chunk = blockIdx.x;
    const int tid   = threadIdx.x;
    const int lane  = tid & 31;
    const int wave  = tid >> 5;

    __shared__ float xbuf[2 * XBUF_FLOATS];          // ~132 KB double buffer
    __shared__ float scorebuf[16];
    __shared__ unsigned short rowlist[ROWS_PER_CHUNK];
    __shared__ int wtot[WAVES];

    // ---- deterministic compaction of unmasked rows (ballot prefix-sum) ----
    const int t0 = chunk * ROWS_PER_CHUNK;
    bool um = (mask[(size_t)b * T_SZ + t0 + tid] == 0);   // mask==True -> -inf -> skip
    unsigned bal = (unsigned)__ballot(um);
    int inwave = __popc(bal & ((1u << lane) - 1u));
    if (lane == 0) wtot[wave] = __popc(bal);
    __syncthreads();
    int prefix = 0, U = 0;
    #pragma unroll
    for (int w2 = 0; w2 < WAVES; ++w2) { if (w2 < wave) prefix += wtot[w2]; U += wtot[w2]; }
    if (um) rowlist[prefix + inwave] = (unsigned short)tid;
    __syncthreads();

    float* part = ws + ((size_t)b * CH + chunk) * PART_STRIDE;

    if (U == 0) {   // fully masked chunk: neutral partial
        for (int i = tid; i < D_SZ; i += THREADS) part[i] = 0.0f;
        if (tid == 0) { part[D_SZ] = -3.0e38f; part[D_SZ + 1] = 0.0f; }
        return;
    }
    const int Upad = (U + 15) & ~15;
    if (tid >= U && tid < Upad) rowlist[tid] = rowlist[0];   // pad with a valid row
    __syncthreads();

    // q slice for the dot phase: lane covers columns lane*4 + j*128
    const float4* q4 = (const float4*)query;
    float4 qv[8];
    #pragma unroll
    for (int j = 0; j < 8; ++j) qv[j] = q4[j * 32 + lane];

    const float* xb = x + ((size_t)b * T_SZ + t0) * D_SZ;    // SADDR base for async loads
    const uint32_t ldsbase = (uint32_t)(uintptr_t)xbuf;

    // issue the 16 async gather rows of `tile` into LDS buffer `buf`
    auto issue_tile = [&](int tile, int buf) {
        const uint32_t bufoff = ldsbase + (uint32_t)(buf * XBUF_FLOATS * 4);
        #pragma unroll 4
        for (int k = 0; k < 16; ++k) {
            uint32_t row  = rowlist[tile * 16 + k];
            uint32_t voff = row * (uint32_t)(D_SZ * 4) + (uint32_t)tid * 16u;
            uint32_t loff = bufoff + (uint32_t)(k * ROWSTRIDE * 4) + (uint32_t)tid * 16u;
            asm volatile("global_load_async_to_lds_b128 %0, %1, %2"
                         :: "v"(loff), "v"(voff), "s"(xb) : "memory");
        }
    };

    float m_run = -3.0e38f, s_run = 0.0f;
    v8f C[CTILES];
    #pragma unroll
    for (int ct = 0; ct < CTILES; ++ct) C[ct] = (v8f){0.f,0.f,0.f,0.f,0.f,0.f,0.f,0.f};

    const int ntiles = Upad >> 4;
    issue_tile(0, 0);                       // prologue: fill buffer 0

    for (int tile = 0; tile < ntiles; ++tile) {
        const int cur = tile & 1;
        if (tile + 1 < ntiles) {
            // buffer cur^1 was released by the barrier at the end of tile-1
            issue_tile(tile + 1, cur ^ 1);
            // async loads complete in order: waiting for <=16 outstanding means
            // this tile's 16 loads (issued first) are all done
            asm volatile("s_wait_asynccnt 16" ::: "memory");
        } else {
            asm volatile("s_wait_asynccnt 0" ::: "memory");
        }
        __syncthreads();
        const float* xtile = xbuf + cur * XBUF_FLOATS;

        // ---- score phase: wave computes dots for tile rows 2*wave, 2*wave+1 ----
        #pragma unroll
        for (int rr = 0; rr < 2; ++rr) {
            const int r = wave * 2 + rr;
            const float* xr = &xtile[r * ROWSTRIDE];
            float dot = 0.0f;
            #pragma unroll
            for (int j = 0; j < 8; ++j) {
                float4 v = *(const float4*)&xr[j * 128 + lane * 4];
                dot += v.x * qv[j].x + v.y * qv[j].y + v.z * qv[j].z + v.w * qv[j].w;
            }
            #pragma unroll
            for (int off = 16; off >= 1; off >>= 1) dot += __shfl_xor(dot, off);
            float score = dot * 0.03125f;                     // 1/sqrt(1024)
            if (tile * 16 + r >= U) score = -1.0e30f;         // padded row -> weight 0
            if (lane == 0) scorebuf[r] = score;
        }
        __syncthreads();

        // ---- online softmax update (identical redundant math in every wave) ----
        float sc = scorebuf[lane & 15];
        float m_tile = sc;
        #pragma unroll
        for (int off = 16; off >= 1; off >>= 1) m_tile = fmaxf(m_tile, __shfl_xor(m_tile, off));
        const float m_new = fmaxf(m_run, m_tile);
        const float scale = __expf(m_run - m_new);
        const float wj    = __expf(sc - m_new);               // weight of tile row (lane&15)
        float wsum = (lane < 16) ? wj : 0.0f;
        #pragma unroll
        for (int off = 16; off >= 1; off >>= 1) wsum += __shfl_xor(wsum, off);
        s_run = s_run * scale + wsum;
        m_run = m_new;
        #pragma unroll
        for (int ct = 0; ct < CTILES; ++ct) C[ct] *= scale;

        // ---- rank-4 weighted accumulate via V_WMMA_F32_16X16X4_F32 ----
        // A (16x4): lanes 0-15 -> K=0 (V0), K=1 (V1); lanes 16-31 -> K=2, K=3. All M rows equal.
        // B (4x16): V0: lanes 0-15 = row K=0, lanes 16-31 = row K=2; V1: K=1 / K=3.
        const int khalf = (lane >> 4) << 1;                   // 0 or 2
        const int nn    = lane & 15;
        #pragma unroll
        for (int g = 0; g < 4; ++g) {
            v2f A;
            A.x = __shfl(wj, g * 4 + khalf);
            A.y = __shfl(wj, g * 4 + khalf + 1);
            const float* r0 = &xtile[(g * 4 + khalf)     * ROWSTRIDE];
            const float* r1 = &xtile[(g * 4 + khalf + 1) * ROWSTRIDE];
            #pragma unroll
            for (int ct = 0; ct < CTILES; ++ct) {
                const int d0 = wave * DSLICE + ct * 16;
                v2f Bv;
                Bv.x = r0[d0 + nn];
                Bv.y = r1[d0 + nn];
                C[ct] = __builtin_amdgcn_wmma_f32_16x16x4_f32(
                            false, A, false, Bv, (short)0, C[ct], false, false);
            }
        }
        __syncthreads();   // current buffer consumed; may be refilled next iteration
    }

    // ---- write partial: C rows are all identical; lanes 0-15, V0 = D[0][n] ----
    if (lane < 16) {
        #pragma unroll
        for (int ct = 0; ct < CTILES; ++ct)
            part[wave * DSLICE + ct * 16 + lane] = C[ct][0];
    }
    if (tid == 0) { part[D_SZ] = m_run; part[D_SZ + 1] = s_run; }
}

__global__ __launch_bounds__(256)
void attnpool_combine(const float* __restrict__ ws, float* __restrict__ out)
{
    const int b   = blockIdx.x;
    const int tid = threadIdx.x;
    const float* base = ws + (size_t)b * CH * PART_STRIDE;

    float m = -3.0e38f;
    #pragma unroll
    for (int c = 0; c < CH; ++c) m = fmaxf(m, base[c * PART_STRIDE + D_SZ]);
    float coef[CH];
    float S = 0.0f;
    #pragma unroll
    for (int c = 0; c < CH; ++c) {
        coef[c] = __expf(base[c * PART_STRIDE + D_SZ] - m);
        S += base[c * PART_STRIDE + D_SZ + 1] * coef[c];
    }
    const float inv = 1.0f / S;

    const int d = tid * 4;
    float4 acc = make_float4(0.f, 0.f, 0.f, 0.f);
    #pragma unroll
    for (int c = 0; c < CH; ++c) {
        float4 v = *(const float4*)&base[c * PART_STRIDE + d];
        acc.x += coef[c] * v.x; acc.y += coef[c] * v.y;
        acc.z += coef[c] * v.z; acc.w += coef[c] * v.w;
    }
    float4 o = make_float4(acc.x * inv, acc.y * inv, acc.z * inv, acc.w * inv);
    *(float4*)&out[(size_t)b * D_SZ + d] = o;
}

extern "C" void kernel_launch(void* const* d_in, const int* in_sizes, int n_in,
                              void* d_out, int out_size, void* d_ws, size_t ws_size,
                              hipStream_t stream)
{
    const float*         x    = (const float*)d_in[0];
    const unsigned char* mask = (const unsigned char*)d_in[1];
    const float*         q    = (const float*)d_in[2];
    float* out = (float*)d_out;
    float* ws  = (float*)d_ws;   // needs 32*16*1032*4 = ~2.1 MB

    dim3 grid1(CH, B_SZ);
    attnpool_partial<<<grid1, THREADS, 0, stream>>>(x, mask, q, ws);
    attnpool_combine<<<B_SZ, 256, 0, stream>>>(ws, out);
}